// ImprovedGCN_69114613730602
// MI455X (gfx1250) — compile-verified
//
#include <hip/hip_runtime.h>
#include <hip/hip_bf16.h>

typedef __attribute__((ext_vector_type(2))) float v2f;
typedef __attribute__((ext_vector_type(8))) float v8f;

// ---------------------------------------------------------------------------
// Stage 0: degree init / accumulate / rsqrt
// ---------------------------------------------------------------------------
__global__ void init_kernel(float* __restrict__ deg, float* __restrict__ psum,
                            float* __restrict__ cnt, int n, int g) {
    int i = blockIdx.x * blockDim.x + threadIdx.x;
    if (i < n) deg[i] = 1.0f;            // self-loop contribution
    if (i < g * 128) psum[i] = 0.0f;
    if (i < g) cnt[i] = 0.0f;
}

__global__ void deg_scatter_kernel(const int* __restrict__ dst,
                                   float* __restrict__ deg, int e) {
    int i = blockIdx.x * blockDim.x + threadIdx.x;
    if (i < e) unsafeAtomicAdd(&deg[dst[i]], 1.0f);
}

__global__ void rsqrt_kernel(float* __restrict__ dinv, int n) {
    int i = blockIdx.x * blockDim.x + threadIdx.x;
    if (i < n) dinv[i] = rsqrtf(dinv[i]);
}

// ---------------------------------------------------------------------------
// Stage 1: WMMA GEMM  hs = (in @ W) * dinv[row]; agg = hs (self-loop seed)
//   in: [nrows,128] f32 row-major, W: [128,128] f32 row-major.
//   W is staged into LDS pre-swizzled so each lane's 16x16x4 B fragment
//   (W[ka][col], W[ka+1][col]) is one contiguous 8-byte ds_load_b64:
//       ldsW[ ((k>>2)*128 + c)*4 + (k&3) ]
//   block = 256 threads = 8 waves; wave computes 16 rows x 128 cols.
// ---------------------------------------------------------------------------
__global__ __launch_bounds__(256) void gemm_wmma_kernel(
    const float* __restrict__ in, const float* __restrict__ W,
    const float* __restrict__ dinv, float* __restrict__ hs,
    float* __restrict__ agg, int nrows) {
    __shared__ float ldsW[128 * 128];  // 64 KB, swizzled

    int tid = threadIdx.x;
    // stage W into LDS with the fragment-friendly swizzle
    {
        const float4* W4 = (const float4*)W;
        for (int i = tid; i < (128 * 128) / 4; i += 256) {
            int k = i >> 5;           // W row (K index)
            int c = (i & 31) * 4;     // starting column of this float4
            float4 w = W4[i];
            int base = (k >> 2) * 512 + (k & 3);
            ldsW[base + (c + 0) * 4] = w.x;
            ldsW[base + (c + 1) * 4] = w.y;
            ldsW[base + (c + 2) * 4] = w.z;
            ldsW[base + (c + 3) * 4] = w.w;
        }
    }
    __syncthreads();

    int wave = tid >> 5;
    int lane = tid & 31;
    int lane16 = lane & 15;
    int khalf = lane >> 4;  // 0: K pair {0,1}, 1: K pair {2,3}
    int rowBase = blockIdx.x * 128 + wave * 16;

    int rowA = rowBase + lane16;
    float amask = (rowA < nrows) ? 1.0f : 0.0f;
    int rowAc = (rowA < nrows) ? rowA : 0;  // clamped, branch-free
    const float* Arow = in + (size_t)rowAc * 128;

    v8f acc[8];
    v8f zero = {0.f, 0.f, 0.f, 0.f, 0.f, 0.f, 0.f, 0.f};
#pragma unroll
    for (int t = 0; t < 8; ++t) acc[t] = zero;

    for (int k0 = 0; k0 < 128; k0 += 4) {
        int ka = k0 + khalf * 2;
        v2f a;
        a.x = Arow[ka + 0] * amask;
        a.y = Arow[ka + 1] * amask;
        const float* Bbase = ldsW + (k0 >> 2) * 512 + khalf * 2;
#pragma unroll
        for (int nt = 0; nt < 8; ++nt) {
            int col = nt * 16 + lane16;
            v2f b = *(const v2f*)(Bbase + col * 4);  // single ds_load_b64
            acc[nt] = __builtin_amdgcn_wmma_f32_16x16x4_f32(
                false, a, false, b, (short)0, acc[nt], false, false);
        }
    }

    // epilogue: scale by dinv[row], write hs and agg (agg seeded with self-loop)
    float dv[8];
    int rows[8];
#pragma unroll
    for (int j = 0; j < 8; ++j) {
        int rr = rowBase + j + khalf * 8;  // C layout: VGPR j holds rows j / j+8
        rows[j] = rr;
        dv[j] = (rr < nrows) ? dinv[rr] : 0.0f;
    }
#pragma unroll
    for (int nt = 0; nt < 8; ++nt) {
        int col = nt * 16 + lane16;
#pragma unroll
        for (int j = 0; j < 8; ++j) {
            int rr = rows[j];
            if (rr < nrows) {
                float val = acc[nt][j] * dv[j];
                size_t o = (size_t)rr * 128 + col;
                hs[o] = val;
                agg[o] = val;
            }
        }
    }
}

// ---------------------------------------------------------------------------
// Stage 2: edge scatter-add. One wave per edge: lane handles 4 contiguous
// floats of the 128-float row (512B contiguous -> coalesced gather).
// ---------------------------------------------------------------------------
__global__ __launch_bounds__(256) void edge_scatter_kernel(
    const int* __restrict__ src, const int* __restrict__ dst,
    const float* __restrict__ hs, float* __restrict__ agg, int e) {
    int gw = (int)((blockIdx.x * blockDim.x + threadIdx.x) >> 5);
    int lane = threadIdx.x & 31;
    if (gw >= e) return;
    int s = src[gw];
    int d = dst[gw];
    const float4 v = *(const float4*)(hs + (size_t)s * 128 + lane * 4);
    float* p = agg + (size_t)d * 128 + lane * 4;
    unsafeAtomicAdd(p + 0, v.x);
    unsafeAtomicAdd(p + 1, v.y);
    unsafeAtomicAdd(p + 2, v.z);
    unsafeAtomicAdd(p + 3, v.w);
}

// ---------------------------------------------------------------------------
// Stage 3: fused  out = relu(BN(agg*dinv + bias))
// one thread per float4
// ---------------------------------------------------------------------------
__global__ void bn_relu_kernel(const float* __restrict__ agg,
                               const float* __restrict__ dinv,
                               const float* __restrict__ bias,
                               const float* __restrict__ gamma,
                               const float* __restrict__ beta,
                               const float* __restrict__ rm,
                               const float* __restrict__ rv,
                               float* __restrict__ out, int n) {
    int idx = blockIdx.x * blockDim.x + threadIdx.x;
    if (idx >= n * 32) return;
    int row = idx >> 5;
    int f = (idx & 31) * 4;
    float dv = dinv[row];
    float4 v = *(const float4*)(agg + (size_t)row * 128 + f);
    float4 o;
    o.x = fmaxf((v.x * dv + bias[f + 0] - rm[f + 0]) * rsqrtf(rv[f + 0] + 1e-5f) * gamma[f + 0] + beta[f + 0], 0.f);
    o.y = fmaxf((v.y * dv + bias[f + 1] - rm[f + 1]) * rsqrtf(rv[f + 1] + 1e-5f) * gamma[f + 1] + beta[f + 1], 0.f);
    o.z = fmaxf((v.z * dv + bias[f + 2] - rm[f + 2]) * rsqrtf(rv[f + 2] + 1e-5f) * gamma[f + 2] + beta[f + 2], 0.f);
    o.w = fmaxf((v.w * dv + bias[f + 3] - rm[f + 3]) * rsqrtf(rv[f + 3] + 1e-5f) * gamma[f + 3] + beta[f + 3], 0.f);
    *(float4*)(out + (size_t)row * 128 + f) = o;
}

// ---------------------------------------------------------------------------
// Stage 4: hierarchical segment-sum pool. LDS 64x128 partial table per block
// (32KB of the 320KB/WGP), flushed once with global f32 atomics.
// ---------------------------------------------------------------------------
__global__ __launch_bounds__(256) void pool_kernel(
    const float* __restrict__ h, const int* __restrict__ batch,
    float* __restrict__ psum, float* __restrict__ cnt, int n) {
    __shared__ float lp[64 * 128];
    __shared__ float lc[64];
    int tid = threadIdx.x;
    for (int i = tid; i < 64 * 128; i += 256) lp[i] = 0.0f;
    if (tid < 64) lc[tid] = 0.0f;
    __syncthreads();

    int lane = tid & 31;
    int wave = tid >> 5;
    int chunk = (n + gridDim.x - 1) / (int)gridDim.x;
    int start = blockIdx.x * chunk;
    int end = min(start + chunk, n);
    for (int i = start + wave; i < end; i += 8) {
        int b = batch[i] & 63;
        const float4 v = *(const float4*)(h + (size_t)i * 128 + lane * 4);
        float* p = lp + b * 128 + lane * 4;
        atomicAdd(p + 0, v.x);  // ds_add_f32
        atomicAdd(p + 1, v.y);
        atomicAdd(p + 2, v.z);
        atomicAdd(p + 3, v.w);
        if (lane == 0) atomicAdd(&lc[b], 1.0f);
    }
    __syncthreads();
    for (int i = tid; i < 64 * 128; i += 256) unsafeAtomicAdd(&psum[i], lp[i]);
    if (tid < 64) unsafeAtomicAdd(&cnt[tid], lc[tid]);
}

// ---------------------------------------------------------------------------
// Stage 5: final head  out[g,c] = (psum[g]/max(cnt,1)) . Wf[:,c] + bf[c]
// ---------------------------------------------------------------------------
__global__ void final_kernel(const float* __restrict__ psum,
                             const float* __restrict__ cnt,
                             const float* __restrict__ Wf,
                             const float* __restrict__ bf,
                             float* __restrict__ out, int g, int c) {
    int tid = blockIdx.x * blockDim.x + threadIdx.x;
    if (tid >= g * c) return;
    int gi = tid / c;
    int ci = tid % c;
    float inv = 1.0f / fmaxf(cnt[gi], 1.0f);
    float s = 0.0f;
    for (int k = 0; k < 128; ++k) s += psum[gi * 128 + k] * inv * Wf[k * c + ci];
    out[tid] = s + bf[ci];
}

// ---------------------------------------------------------------------------
extern "C" void kernel_launch(void* const* d_in, const int* in_sizes, int n_in,
                              void* d_out, int out_size, void* d_ws, size_t ws_size,
                              hipStream_t stream) {
    const float* x   = (const float*)d_in[0];
    const int* ei    = (const int*)d_in[1];
    const int* batch = (const int*)d_in[2];
    const float* W1  = (const float*)d_in[3];
    const float* b1  = (const float*)d_in[4];
    const float* g1  = (const float*)d_in[5];
    const float* be1 = (const float*)d_in[6];
    const float* rm1 = (const float*)d_in[7];
    const float* rv1 = (const float*)d_in[8];
    const float* W2  = (const float*)d_in[9];
    const float* b2  = (const float*)d_in[10];
    const float* g2  = (const float*)d_in[11];
    const float* be2 = (const float*)d_in[12];
    const float* rm2 = (const float*)d_in[13];
    const float* rv2 = (const float*)d_in[14];
    const float* Wf  = (const float*)d_in[15];
    const float* bf  = (const float*)d_in[16];

    int N = in_sizes[0] / 128;
    int E = in_sizes[1] / 2;
    int C = in_sizes[16];
    int G = out_size / C;
    const int* src = ei;
    const int* dst = ei + E;

    float* ws = (float*)d_ws;
    int Npad = (N + 3) & ~3;
    float* dinv = ws;                                   // [N]   (deg then rsqrt in place)
    float* bufA = dinv + Npad;                          // [N,128] layer output
    float* bufB = bufA + (size_t)N * 128;               // [N,128] agg
    float* bufC = bufB + (size_t)N * 128;               // [N,128] hs (pre-scaled h)
    float* psum = bufC + (size_t)N * 128;               // [G,128]
    float* cnt  = psum + (size_t)G * 128;               // [G]

    int t256 = 256;
    // degree / norm prep + zero pool accumulators (every call: deterministic)
    init_kernel<<<(N + 255) / 256, t256, 0, stream>>>(dinv, psum, cnt, N, G);
    deg_scatter_kernel<<<(E + 255) / 256, t256, 0, stream>>>(dst, dinv, E);
    rsqrt_kernel<<<(N + 255) / 256, t256, 0, stream>>>(dinv, N);

    int gemmBlocks = (N + 127) / 128;
    int edgeBlocks = (E + 7) / 8;           // 8 waves/block, 1 wave/edge
    int bnBlocks   = (N * 32 + 255) / 256;

    // ---- layer 1
    gemm_wmma_kernel<<<gemmBlocks, t256, 0, stream>>>(x, W1, dinv, bufC, bufB, N);
    edge_scatter_kernel<<<edgeBlocks, t256, 0, stream>>>(src, dst, bufC, bufB, E);
    bn_relu_kernel<<<bnBlocks, t256, 0, stream>>>(bufB, dinv, b1, g1, be1, rm1, rv1, bufA, N);

    // ---- layer 2
    gemm_wmma_kernel<<<gemmBlocks, t256, 0, stream>>>(bufA, W2, dinv, bufC, bufB, N);
    edge_scatter_kernel<<<edgeBlocks, t256, 0, stream>>>(src, dst, bufC, bufB, E);
    bn_relu_kernel<<<bnBlocks, t256, 0, stream>>>(bufB, dinv, b2, g2, be2, rm2, rv2, bufA, N);

    // ---- pool + head
    pool_kernel<<<512, t256, 0, stream>>>(bufA, batch, psum, cnt, N);
    final_kernel<<<(G * C + 255) / 256, t256, 0, stream>>>(psum, cnt, Wf, bf, (float*)d_out, G, C);
}